// DetectPeaks_69552700391737
// MI455X (gfx1250) — compile-verified
//
#include <hip/hip_runtime.h>
#include <stdint.h>

#define Hh 1024
#define Ww 1024
#define TX 64
#define TY 16
#define LW 66   // TX + 2 halo
#define LH 18   // TY + 2 halo
#define THRESH 0.1f

typedef unsigned int v4u __attribute__((ext_vector_type(4)));
typedef int          v8i __attribute__((ext_vector_type(8)));
typedef int          v4i __attribute__((ext_vector_type(4)));

// Order-preserving float -> uint key (handles negatives; inputs are [0,1) anyway)
__device__ __forceinline__ unsigned fkey(float f) {
    unsigned u = __float_as_uint(f);
    return (u & 0x80000000u) ? ~u : (u | 0x80000000u);
}
__device__ __forceinline__ float fdec(unsigned k) {
    unsigned u = (k & 0x80000000u) ? (k ^ 0x80000000u) : ~k;
    return __uint_as_float(u);
}
__device__ __forceinline__ unsigned umin2(unsigned a, unsigned b) { return a < b ? a : b; }
__device__ __forceinline__ unsigned umax2(unsigned a, unsigned b) { return a > b ? a : b; }

// ---------------------------------------------------------------------------
// Peak-detection stencil: TDM tile load -> LDS, 3x3 max windows from LDS.
// Defined FIRST so the disasm snippet shows the tensor_load_to_lds encoding.
// ---------------------------------------------------------------------------
__global__ __launch_bounds__(256) void peaks_kernel(const float* __restrict__ hmap,
                                                    const unsigned* __restrict__ ws,
                                                    int* __restrict__ out) {
    __shared__ float tile[LH][LW];

    const int x0 = blockIdx.x * TX;
    const int y0 = blockIdx.y * TY;
    const int bz = blockIdx.z;

    // Clamped fixed-size window [ys, ys+LH) x [xs, xs+LW): always fully in-bounds.
    int ys = y0 - 1; if (ys < 0) ys = 0; if (ys > Hh - LH) ys = Hh - LH;
    int xs = x0 - 1; if (xs < 0) xs = 0; if (xs > Ww - LW) xs = Ww - LW;

    // ---- Tensor Data Mover: DMA the 18x66 f32 tile into LDS (wave 0 issues) ----
    const uint64_t gaddr =
        (uint64_t)(uintptr_t)(hmap + (size_t)bz * Hh * Ww + (size_t)ys * Ww + xs);
    unsigned lds_off = (unsigned)(uintptr_t)&tile[0][0]; // low 32 bits of flat LDS addr = LDS byte offset
    lds_off = (unsigned)__builtin_amdgcn_readfirstlane((int)lds_off);

    if (threadIdx.x < 32) {
        v4u g0;
        g0.x = 1u;                                              // count=1, user descriptor
        g0.y = lds_off;                                         // lds_addr
        g0.z = (unsigned)(gaddr & 0xFFFFFFFFu);                 // global_addr[31:0]
        g0.w = (unsigned)((gaddr >> 32) & 0x01FFFFFFu) | (2u << 30); // addr[56:32] | type=2
        v8i g1;
        g1[0] = (int)(2u << 16);                                // data_size=2 (4 bytes)
        g1[1] = (int)((unsigned)LW << 16);                      // tensor_dim0 lo16
        g1[2] = (int)((unsigned)LH << 16);                      // dim0 hi16=0 | tensor_dim1 lo16
        g1[3] = (int)((unsigned)LW << 16);                      // dim1 hi16=0 | tile_dim0=66
        g1[4] = (int)((unsigned)LH);                            // tile_dim1=18, tile_dim2=0
        g1[5] = (int)Ww;                                        // tensor_dim0_stride lo32 = 1024
        g1[6] = 0;                                              // stride hi16 | dim1_stride lo16
        g1[7] = 0;                                              // dim1_stride hi32
        v4i g2 = {0, 0, 0, 0};
        v4i g3 = {0, 0, 0, 0};
        v8i gx2 = {0, 0, 0, 0, 0, 0, 0, 0};                     // unused extra group (6-arg form)
        __builtin_amdgcn_tensor_load_to_lds(g0, g1, g2, g3, gx2, 0);
        __builtin_amdgcn_s_wait_tensorcnt(0);
    }
    __syncthreads();

    const float hmin = fdec(ws[0]);
    const float hmax = fdec(ws[1]);
    const float inv  = 1.0f / (hmax - hmin);

    const int lx = threadIdx.x & (TX - 1);
    const int ry = threadIdx.x >> 6; // 0..3
    const int gx = x0 + lx;
    const int c  = gx - xs;
    const bool xb0 = (gx == 0);
    const bool xb1 = (gx == Ww - 1);

    int* outz = out + (size_t)bz * Hh * Ww;

#pragma unroll
    for (int k = 0; k < 4; ++k) {
        const int ly = k * 4 + ry;
        const int gy = y0 + ly;
        const int r  = gy - ys;

        const float v   = tile[r][c];
        const float n_v = (v - hmin) * inv;
        const float hmv = (n_v < THRESH) ? 0.0f : n_v;

        int result;
        if (gy == 0 || gy == Hh - 1) {
            // lm row is zero-padded -> local_max = (hm==0); eroded = true; xor = (hm != 0)
            result = (hmv != 0.0f) ? 1 : 0;
        } else {
            // reflect pad in width: window center col = 2 at x=0, W-3 at x=W-1, else x
            int cc = c;
            if (xb0) cc = 2 - xs;
            if (xb1) cc = (Ww - 3) - xs;
            float m = tile[r - 1][cc - 1];
            m = fmaxf(m, tile[r - 1][cc    ]);
            m = fmaxf(m, tile[r - 1][cc + 1]);
            m = fmaxf(m, tile[r    ][cc - 1]);
            m = fmaxf(m, tile[r    ][cc    ]);
            m = fmaxf(m, tile[r    ][cc + 1]);
            m = fmaxf(m, tile[r + 1][cc - 1]);
            m = fmaxf(m, tile[r + 1][cc    ]);
            m = fmaxf(m, tile[r + 1][cc + 1]);
            const float n_m = (m - hmin) * inv;          // g is monotone: max(g(v)) = g(max v)
            const float lmv = (n_m < THRESH) ? 0.0f : n_m;
            const bool  lmax = (lmv == hmv);
            const bool  er   = (xb0 || xb1) ? true : (n_m < THRESH);
            result = (lmax != er) ? 1 : 0;
        }
        outz[(size_t)gy * Ww + gx] = result;
    }
}

__global__ void init_ws_kernel(unsigned* ws) {
    ws[0] = 0xFFFFFFFFu; // running min key
    ws[1] = 0u;          // running max key
}

__global__ __launch_bounds__(256) void minmax_kernel(const float4* __restrict__ in,
                                                     unsigned* __restrict__ ws, int n4) {
    unsigned kmin = 0xFFFFFFFFu, kmax = 0u;
    int stride = gridDim.x * blockDim.x;
    for (int i = blockIdx.x * blockDim.x + threadIdx.x; i < n4; i += stride) {
        float4 v = in[i];
        unsigned a = fkey(v.x), b = fkey(v.y), c = fkey(v.z), d = fkey(v.w);
        kmin = umin2(kmin, umin2(umin2(a, b), umin2(c, d)));
        kmax = umax2(kmax, umax2(umax2(a, b), umax2(c, d)));
    }
    __shared__ unsigned smin[256];
    __shared__ unsigned smax[256];
    int t = threadIdx.x;
    smin[t] = kmin; smax[t] = kmax;
    __syncthreads();
    for (int s = 128; s > 0; s >>= 1) {
        if (t < s) {
            smin[t] = umin2(smin[t], smin[t + s]);
            smax[t] = umax2(smax[t], smax[t + s]);
        }
        __syncthreads();
    }
    if (t == 0) {
        atomicMin(&ws[0], smin[0]);
        atomicMax(&ws[1], smax[0]);
    }
}

extern "C" void kernel_launch(void* const* d_in, const int* in_sizes, int n_in,
                              void* d_out, int out_size, void* d_ws, size_t ws_size,
                              hipStream_t stream) {
    const float* hmap = (const float*)d_in[0];
    unsigned* ws = (unsigned*)d_ws;
    int* out = (int*)d_out;

    const int n  = in_sizes[0];          // B*C*H*W = 33554432
    const int n4 = n / 4;
    const int bc = n / (Hh * Ww);        // B*C = 32

    init_ws_kernel<<<dim3(1), dim3(1), 0, stream>>>(ws);
    minmax_kernel<<<dim3(2048), dim3(256), 0, stream>>>((const float4*)hmap, ws, n4);

    dim3 grid(Ww / TX, Hh / TY, bc);     // 16 x 64 x 32
    peaks_kernel<<<grid, dim3(256), 0, stream>>>(hmap, ws, out);
}